// Encoder_23922967839341
// MI455X (gfx1250) — compile-verified
//
#include <hip/hip_runtime.h>
#include <hip/hip_bf16.h>
#include <math.h>

// Problem constants (fixed by the reference)
#define B_ 256
#define T_ 256
#define N_ 128
#define M_ 256

typedef __attribute__((ext_vector_type(2))) float v2f;
typedef __attribute__((ext_vector_type(8))) float v8f;

// exact tanh via v_exp_f32; clamp so e^{2x} never overflows to inf (tanh(10)==1 in f32)
__device__ __forceinline__ float fast_tanh(float x) {
    float xc = fminf(fmaxf(x, -10.f), 10.f);
    float e2 = __expf(2.f * xc);
    return (e2 - 1.f) / (e2 + 1.f);
}
__device__ __forceinline__ float fast_sigmoid(float x) {
    return 1.f / (1.f + __expf(-x));
}

// wave32 xor-butterfly reductions via ds_swizzle (group-of-32: and=0x1f, xor=k)
__device__ __forceinline__ float wave32_max(float x) {
    x = fmaxf(x, __int_as_float(__builtin_amdgcn_ds_swizzle(__float_as_int(x), 0x401f)));
    x = fmaxf(x, __int_as_float(__builtin_amdgcn_ds_swizzle(__float_as_int(x), 0x201f)));
    x = fmaxf(x, __int_as_float(__builtin_amdgcn_ds_swizzle(__float_as_int(x), 0x101f)));
    x = fmaxf(x, __int_as_float(__builtin_amdgcn_ds_swizzle(__float_as_int(x), 0x081f)));
    x = fmaxf(x, __int_as_float(__builtin_amdgcn_ds_swizzle(__float_as_int(x), 0x041f)));
    return x;
}
__device__ __forceinline__ float wave32_sum(float x) {
    x += __int_as_float(__builtin_amdgcn_ds_swizzle(__float_as_int(x), 0x401f));
    x += __int_as_float(__builtin_amdgcn_ds_swizzle(__float_as_int(x), 0x201f));
    x += __int_as_float(__builtin_amdgcn_ds_swizzle(__float_as_int(x), 0x101f));
    x += __int_as_float(__builtin_amdgcn_ds_swizzle(__float_as_int(x), 0x081f));
    x += __int_as_float(__builtin_amdgcn_ds_swizzle(__float_as_int(x), 0x041f));
    return x;
}

// ---------------------------------------------------------------------------
// Kernel 1: Ud[b][n][t] = sum_k data[b][k][n] * U_e[t][k]
// Batched GEMM per b: (N=128 x K=256) x (K=256 x T=256) using f32 WMMA.
// One wave computes a 16x32 output block (two 16x16 tiles sharing the A frag).
// A fragment (16x4 f32): lanes 0-15 hold K=k,k+1 ; lanes 16-31 hold K=k+2,k+3.
// B fragment (4x16 f32): mirrored layout over the N(=t) dimension.
// D (16x16 f32, 8 VGPRs): vgpr r = element (M = r + 8*half, N = lane&15).
// ---------------------------------------------------------------------------
__global__ __launch_bounds__(256) void ud_gemm_kernel(
    const float* __restrict__ data, const float* __restrict__ U_e,
    float* __restrict__ Ud) {
    const int wave = threadIdx.x >> 5;
    const int lane = threadIdx.x & 31;
    const int pair = blockIdx.x * 8 + wave;   // 16384 tile-pairs total
    const int bb   = pair >> 6;               // 64 pairs per batch element
    const int rem  = pair & 63;
    const int n0   = (rem >> 3) << 4;         // 8 n-tiles (16 rows each)
    const int t0   = (rem & 7) << 5;          // 8 t-pairs (32 cols each)
    const int half = lane >> 4;
    const int l    = lane & 15;

    // A[n][k] = data[b][k][n]  (n fast dim -> lanes 0..15 coalesce over n)
    const float* Arow  = data + (size_t)bb * (T_ * N_) + n0 + l;
    // B[k][t] = U_e[t][k]      (k fast dim -> contiguous per lane)
    const float* Brow0 = U_e + (size_t)(t0 + l) * T_;
    const float* Brow1 = U_e + (size_t)(t0 + 16 + l) * T_;

    v8f acc0 = {0.f, 0.f, 0.f, 0.f, 0.f, 0.f, 0.f, 0.f};
    v8f acc1 = {0.f, 0.f, 0.f, 0.f, 0.f, 0.f, 0.f, 0.f};
    #pragma unroll 4
    for (int k = 0; k < T_; k += 4) {
        const int ka = k + 2 * half;
        v2f a, b0, b1;
        a.x  = Arow[(size_t)ka * N_];
        a.y  = Arow[(size_t)(ka + 1) * N_];
        b0.x = Brow0[ka];
        b0.y = Brow0[ka + 1];
        b1.x = Brow1[ka];
        b1.y = Brow1[ka + 1];
        // 8 args: (neg_a, A, neg_b, B, c_mod, C, reuse_a, reuse_b)
        acc0 = __builtin_amdgcn_wmma_f32_16x16x4_f32(
            false, a, false, b0, (short)0, acc0, false, false);
        acc1 = __builtin_amdgcn_wmma_f32_16x16x4_f32(
            false, a, false, b1, (short)0, acc1, false, false);
    }

    // Store D -> Ud[b][n0+M][t0(+16)+l]  (lanes 0..15 contiguous over t)
    float* Crow = Ud + ((size_t)bb * N_ + n0 + 8 * half) * T_ + t0 + l;
    #pragma unroll
    for (int r = 0; r < 8; ++r) {
        Crow[(size_t)r * T_]      = acc0[r];
        Crow[(size_t)r * T_ + 16] = acc1[r];
    }
}

// ---------------------------------------------------------------------------
// Kernel 2: the sequential recurrence. One workgroup per batch element b,
// 256 threads, full T-loop in-kernel. h/c live in LDS; no global sync.
// ---------------------------------------------------------------------------
__global__ __launch_bounds__(256) void encoder_recurrence_kernel(
    const float* __restrict__ data, const float* __restrict__ h0,
    const float* __restrict__ s0,   const float* __restrict__ W_e,
    const float* __restrict__ v_e,  const float* __restrict__ W_ih,
    const float* __restrict__ W_hh, const float* __restrict__ b_ih,
    const float* __restrict__ b_hh, const float* __restrict__ Ud,
    float* __restrict__ outH, float* __restrict__ outA) {
    __shared__ __align__(16) float q_s[2 * M_];   // [h | c]
    __shared__ __align__(16) float Wq_s[T_];
    __shared__ __align__(16) float v_s[T_];
    __shared__ __align__(16) float bias_s[4 * M_];
    __shared__ __align__(16) float epart_s[256];
    __shared__ __align__(16) float red_s[8];      // [0..3]=wave max, [4..7]=wave sum
    __shared__ __align__(16) float xt_s[N_];

    const int tid = threadIdx.x;
    const int bb  = blockIdx.x;

    // init: state, combined bias, v
    q_s[tid]      = h0[(size_t)bb * M_ + tid];
    q_s[M_ + tid] = s0[(size_t)bb * M_ + tid];
    v_s[tid]      = v_e[tid];
    #pragma unroll
    for (int g = 0; g < 4; ++g)
        bias_s[g * M_ + tid] = b_ih[g * M_ + tid] + b_hh[g * M_ + tid];
    __syncthreads();

    const int n_id = tid & (N_ - 1);
    const int half = tid >> 7;
    const float* ud_thread = Ud + ((size_t)bb * N_ + n_id) * T_ + half * (T_ / 2);

    for (int t = 0; t < T_; ++t) {
        // ---- phase 1: Wq[tq] = W_e[tq,:] . [h|c]   (tq = tid)
        {
            const float4* Wrow = (const float4*)(W_e + (size_t)tid * (2 * M_));
            const float4* qv   = (const float4*)q_s;
            float acc = 0.f;
            #pragma unroll 8
            for (int j = 0; j < (2 * M_) / 4; ++j) {
                float4 w = Wrow[j];
                float4 q = qv[j];
                acc += w.x * q.x + w.y * q.y + w.z * q.z + w.w * q.w;
            }
            Wq_s[tid] = acc;
        }
        __syncthreads();

        // ---- phase 2: two threads per n, each sums half the T range
        {
            float acc = 0.f;
            const int tbase = half * (T_ / 2);
            #pragma unroll 4
            for (int k = 0; k < T_ / 2; ++k) {
                float u = ud_thread[k];
                acc += v_s[tbase + k] * fast_tanh(Wq_s[tbase + k] + u);
            }
            epart_s[tid] = acc;
        }
        __syncthreads();

        // ---- phase 3: softmax over n (threads < 128 hold a value);
        //      wave32 ds_swizzle butterflies + tiny LDS cross-wave combine
        const bool live = (tid < N_);
        const float e_n = live ? (epart_s[tid] + epart_s[tid + N_]) : 0.f;

        float wmax = wave32_max(live ? e_n : -3.0e38f);
        if (live && (tid & 31) == 0) red_s[tid >> 5] = wmax;
        __syncthreads();
        const float mx = fmaxf(fmaxf(red_s[0], red_s[1]), fmaxf(red_s[2], red_s[3]));

        const float p = live ? __expf(e_n - mx) : 0.f;
        float wsum = wave32_sum(p);
        if (live && (tid & 31) == 0) red_s[4 + (tid >> 5)] = wsum;
        __syncthreads();
        const float inv_sum = 1.f / (red_s[4] + red_s[5] + red_s[6] + red_s[7]);

        if (live) {
            const float alpha = p * inv_sum;
            const float x_t   = data[((size_t)bb * T_ + t) * N_ + tid];
            xt_s[tid] = alpha * x_t;
            outA[((size_t)bb * T_ + t) * N_ + tid] = alpha;
        }
        __syncthreads();

        // ---- phase 4: gates; thread m computes rows m, 256+m, 512+m, 768+m
        float acc0 = 0.f, acc1 = 0.f, acc2 = 0.f, acc3 = 0.f;
        {
            const float4* xv = (const float4*)xt_s;
            const float4* w0 = (const float4*)(W_ih + (size_t)(0 * M_ + tid) * N_);
            const float4* w1 = (const float4*)(W_ih + (size_t)(1 * M_ + tid) * N_);
            const float4* w2 = (const float4*)(W_ih + (size_t)(2 * M_ + tid) * N_);
            const float4* w3 = (const float4*)(W_ih + (size_t)(3 * M_ + tid) * N_);
            #pragma unroll 4
            for (int k = 0; k < N_ / 4; ++k) {
                const float4 x = xv[k];
                float4 a;
                a = w0[k]; acc0 += a.x * x.x + a.y * x.y + a.z * x.z + a.w * x.w;
                a = w1[k]; acc1 += a.x * x.x + a.y * x.y + a.z * x.z + a.w * x.w;
                a = w2[k]; acc2 += a.x * x.x + a.y * x.y + a.z * x.z + a.w * x.w;
                a = w3[k]; acc3 += a.x * x.x + a.y * x.y + a.z * x.z + a.w * x.w;
            }
            const float4* hv = (const float4*)q_s;  // h part
            const float4* u0 = (const float4*)(W_hh + (size_t)(0 * M_ + tid) * M_);
            const float4* u1 = (const float4*)(W_hh + (size_t)(1 * M_ + tid) * M_);
            const float4* u2 = (const float4*)(W_hh + (size_t)(2 * M_ + tid) * M_);
            const float4* u3 = (const float4*)(W_hh + (size_t)(3 * M_ + tid) * M_);
            #pragma unroll 4
            for (int k = 0; k < M_ / 4; ++k) {
                const float4 h = hv[k];
                float4 a;
                a = u0[k]; acc0 += a.x * h.x + a.y * h.y + a.z * h.z + a.w * h.w;
                a = u1[k]; acc1 += a.x * h.x + a.y * h.y + a.z * h.z + a.w * h.w;
                a = u2[k]; acc2 += a.x * h.x + a.y * h.y + a.z * h.z + a.w * h.w;
                a = u3[k]; acc3 += a.x * h.x + a.y * h.y + a.z * h.z + a.w * h.w;
            }
        }
        __syncthreads();  // everyone done reading q_s before state update

        // ---- phase 5: LSTM cell (PyTorch gate order i,f,g,o), m = tid
        {
            const float iv = acc0 + bias_s[tid];
            const float fv = acc1 + bias_s[M_ + tid];
            const float gv = acc2 + bias_s[2 * M_ + tid];
            const float ov = acc3 + bias_s[3 * M_ + tid];
            const float c_old = q_s[M_ + tid];
            const float c_new = fast_sigmoid(fv) * c_old + fast_sigmoid(iv) * fast_tanh(gv);
            const float h_new = fast_sigmoid(ov) * fast_tanh(c_new);
            q_s[tid]      = h_new;
            q_s[M_ + tid] = c_new;
            outH[((size_t)bb * T_ + t) * M_ + tid] = h_new;
        }
        __syncthreads();
    }
}

extern "C" void kernel_launch(void* const* d_in, const int* in_sizes, int n_in,
                              void* d_out, int out_size, void* d_ws, size_t ws_size,
                              hipStream_t stream) {
    (void)in_sizes; (void)n_in; (void)out_size; (void)ws_size;

    const float* data = (const float*)d_in[0];
    const float* h0   = (const float*)d_in[1];
    const float* s0   = (const float*)d_in[2];
    const float* W_e  = (const float*)d_in[3];
    const float* U_e  = (const float*)d_in[4];
    const float* v_e  = (const float*)d_in[5];
    const float* W_ih = (const float*)d_in[6];
    const float* W_hh = (const float*)d_in[7];
    const float* b_ih = (const float*)d_in[8];
    const float* b_hh = (const float*)d_in[9];

    float* outH = (float*)d_out;                         // [B,T,M]
    float* outA = outH + (size_t)B_ * T_ * M_;           // [B,T,N]
    float* Ud   = (float*)d_ws;                          // [B,N,T] = 33.5 MB

    // Phase A: WMMA batched GEMM for the time-invariant attention term.
    // 16384 two-tile (16x32) blocks, 8 waves/block -> 2048 blocks.
    ud_gemm_kernel<<<dim3((B_ * (N_ / 16) * (T_ / 32)) / 8), dim3(256), 0, stream>>>(
        data, U_e, Ud);

    // Phase B: sequential recurrence, one WG per batch element.
    encoder_recurrence_kernel<<<dim3(B_), dim3(256), 0, stream>>>(
        data, h0, s0, W_e, v_e, W_ih, W_hh, b_ih, b_hh, Ud, outH, outA);
}